// DifferentiableHexagonalSensor_83133386982140
// MI455X (gfx1250) — compile-verified
//
#include <hip/hip_runtime.h>
#include <math.h>

typedef __attribute__((ext_vector_type(2))) float v2f;
typedef __attribute__((ext_vector_type(4))) float v4f;
typedef __attribute__((ext_vector_type(8))) float v8f;

#define NPTS     8388608
#define NPIX     1801
#define PSTRIDE  1808      // padded row stride, multiple of 16
#define THREADS  256
#define MAXBLK   1024

// Closed form of the reference lookup table: pixel index of axial (q, r) is
// hex_row_base(q) + r for all valid (|q|<=24, |r|<=24, |q+r|<=24) cells.
__device__ __forceinline__ int hex_row_base(int q) {
    int t = q + 24;                                       // 0..48
    int pneg = 25 * t + ((t * (t - 1)) >> 1) + 24 + q;    // q <= 0
    int ppos = 949 + 49 * (q - 1) - ((q * (q - 1)) >> 1); // q >  0
    return (q <= 0) ? pneg : ppos;
}

__global__ __launch_bounds__(THREADS) void hex_splat(
    const float* __restrict__ xs, const float* __restrict__ ys,
    const float* __restrict__ vs, float* __restrict__ partials, int nblk)
{
    __shared__ float hist[PSTRIDE];
    for (int i = threadIdx.x; i < PSTRIDE; i += THREADS) hist[i] = 0.0f;
    __syncthreads();

    // 7-neighbor stencil (KERNEL_RINGS=1), hex_size=1, rot=0, offset=(0,0)
    const float nox[7] = {-1.7320508f, -0.8660254f, -0.8660254f, 0.0f,
                           0.8660254f,  0.8660254f,  1.7320508f};
    const float noy[7] = { 0.0f, 1.5f, -1.5f, 0.0f, 1.5f, -1.5f, 0.0f};
    const int   na[7]  = {-1, -1, 0, 0, 0, 1, 1};
    const int   nb[7]  = { 0,  1,-1, 0, 1,-1, 0};

    const int nvec = NPTS / 4;
    const v4f* __restrict__ x4 = (const v4f*)xs;
    const v4f* __restrict__ y4 = (const v4f*)ys;
    const v4f* __restrict__ v4 = (const v4f*)vs;

    for (int i = blockIdx.x * THREADS + threadIdx.x; i < nvec;
         i += nblk * THREADS) {
        v4f px = __builtin_nontemporal_load(&x4[i]);
        v4f py = __builtin_nontemporal_load(&y4[i]);
        v4f pv = __builtin_nontemporal_load(&v4[i]);
#pragma unroll
        for (int j = 0; j < 4; ++j) {
            float x = px[j], y = py[j], v = pv[j];
            // cartesian -> fractional axial (hex_size = 1)
            float q = 0.57735026f * x - 0.33333334f * y;
            float r = 0.66666669f * y;
            // axial round (round-half-even matches jnp.round)
            float s0 = -q - r;
            float qi = rintf(q), ri = rintf(r), si = rintf(s0);
            float dq = fabsf(qi - q), dr = fabsf(ri - r), dd = fabsf(si - s0);
            float qb_f = (dq > dr && dq > dd) ? (-ri - si) : qi;
            float rb_f = (dr > dq && dr > dd) ? (-qi - si) : ri;
            int   qb = (int)qb_f, rb = (int)rb_f;
            float dx = x - 1.7320508f * (qb_f + 0.5f * rb_f);
            float dy = y - 1.5f * rb_f;

            float w[7];
            float wsum = 0.0f;
#pragma unroll
            for (int n = 0; n < 7; ++n) {
                float ax = fabsf(dx - nox[n]);
                float ay = fabsf(dy - noy[n]);
                float m  = fmaxf(ay, 0.8660254f * ax + 0.5f * ay);
                // exp(-0.5*((m/inr)/0.5)^2) = 2^(-(8/3)*log2(e) * m^2)
                w[n] = exp2f(-3.8471868f * m * m);
                wsum += w[n];
            }
            float scale = v / wsum;

            int rowb[3];
#pragma unroll
            for (int t = 0; t < 3; ++t) rowb[t] = hex_row_base(qb - 1 + t);

#pragma unroll
            for (int n = 0; n < 7; ++n) {
                int qq = qb + na[n];
                int rr = rb + nb[n];
                int ss = qq + rr;
                bool valid = (qq >= -24) & (qq <= 24) & (rr >= -24) &
                             (rr <= 24) & (ss >= -24) & (ss <= 24);
                if (valid)
                    atomicAdd(&hist[rowb[na[n] + 1] + rr], scale * w[n]);
            }
        }
    }
    __syncthreads();
    float* row = partials + (size_t)blockIdx.x * PSTRIDE;
    for (int i = threadIdx.x; i < PSTRIDE; i += THREADS) row[i] = hist[i];
}

// Cross-block reduction: out[c] = sum_b P[b][c], done as ones(16x4) x P(4x16)
// accumulated with V_WMMA_F32_16X16X4_F32. A = ones makes the result invariant
// to the K-slot layout of B; columns map to lane%16 per the C/D layout.
__global__ __launch_bounds__(32) void hex_reduce(
    const float* __restrict__ partials, float* __restrict__ out, int nblk)
{
    int lane = threadIdx.x;
    int col  = blockIdx.x * 16 + (lane & 15);
    int half = lane >> 4;   // 0 for lanes 0-15, 1 for lanes 16-31

    v2f a; a[0] = 1.0f; a[1] = 1.0f;
    v8f c = {};
    const float* p = partials + col;
    for (int kb = 0; kb < nblk; kb += 4) {
        v2f b;
        b[0] = p[(size_t)(kb + half)     * PSTRIDE];
        b[1] = p[(size_t)(kb + half + 2) * PSTRIDE];
        c = __builtin_amdgcn_wmma_f32_16x16x4_f32(
                /*neg_a=*/false, a, /*neg_b=*/false, b,
                /*c_mod=*/(short)0, c, /*reuse_a=*/false, /*reuse_b=*/false);
    }
    if (lane < 16 && col < NPIX) out[col] = c[0];
}

extern "C" void kernel_launch(void* const* d_in, const int* in_sizes, int n_in,
                              void* d_out, int out_size, void* d_ws, size_t ws_size,
                              hipStream_t stream) {
    const float* x = (const float*)d_in[0];
    const float* y = (const float*)d_in[1];
    const float* v = (const float*)d_in[2];
    float* out = (float*)d_out;
    float* ws  = (float*)d_ws;

    int nblk = MAXBLK;
    size_t need = (size_t)nblk * PSTRIDE * sizeof(float);
    if (ws_size < need) {
        nblk = (int)(ws_size / (PSTRIDE * sizeof(float)));
        nblk &= ~3;               // keep a multiple of 4 for the WMMA reduce
        if (nblk < 4) nblk = 4;
    }

    hex_splat<<<nblk, THREADS, 0, stream>>>(x, y, v, ws, nblk);
    hex_reduce<<<(NPIX + 15) / 16, 32, 0, stream>>>(ws, out, nblk);
}